// MultiHeadGATSkipLayer_4595615006841
// MI455X (gfx1250) — compile-verified
//
#include <hip/hip_runtime.h>
#include <math.h>

// ---------------------------------------------------------------------------
// MultiHeadGAT + skip on gfx1250 (MI455X, wave32, WMMA + async-to-LDS)
//   N=50000 nodes, E=800000 edges, IN=256, HEADS=4, C=64  (H*C = 256)
// ---------------------------------------------------------------------------

typedef __attribute__((ext_vector_type(2))) float v2f;
typedef __attribute__((ext_vector_type(8))) float v8f;

#define IN_DIM   256
#define OUTF     256      // HEADS * OUT_DIM
#define HEADS    4
#define NEG_SLOPE 0.2f
#define EPS_      1e-16f

// ---- GEMM tiling -----------------------------------------------------------
#define BM 32
#define BN 64
#define KC 32
#define NCHUNK (IN_DIM / KC)   // 8
#define AS_STRIDE 36      // 144B row: 16B-aligned chunks, conflict-free frag reads
#define BS_STRIDE 68      // 272B row: 16B-aligned chunks

// async copy of 16B/lane straight into LDS, tracked by ASYNCcnt
#define ASYNC_B128(ldsaddr, gptr)                                   \
  asm volatile("global_load_async_to_lds_b128 %0, %1, off"          \
               :: "v"(ldsaddr), "v"(gptr) : "memory")

// out[M,256] = x[M,256] @ Wm[256,256]; blockIdx.z picks (W -> h) or (skip_W -> skip)
__global__ __launch_bounds__(256) void gat_gemm_wmma(
    const float* __restrict__ x, const float* __restrict__ W0,
    const float* __restrict__ W1, float* __restrict__ out0,
    float* __restrict__ out1, int M)
{
  const float* Wm   = (blockIdx.z == 0) ? W0 : W1;
  float*       outp = (blockIdx.z == 0) ? out0 : out1;

  __shared__ float As[2][BM * AS_STRIDE];
  __shared__ float Bs[2][KC * BS_STRIDE];

  const int tid  = threadIdx.x;
  const int m0   = blockIdx.x * BM;
  const int n0   = blockIdx.y * BN;
  const int wave = tid >> 5;
  const int lane = tid & 31;
  const int wr   = wave >> 2;      // 0..1 : 16-row band
  const int wc   = wave & 3;       // 0..3 : 16-col band
  const int half = lane >> 4;      // K-pair select (ISA A/B layout)
  const int l16  = lane & 15;

  // ---- staging coordinates (16B per lane per async instruction) ----
  const int arow  = tid >> 3;            // 0..31
  const int acol4 = (tid & 7) * 4;       // 0..28
  int gra = m0 + arow; if (gra > M - 1) gra = M - 1;   // clamp: keeps async count
                                                       // uniform; bad rows discarded
  const int brow0 = tid >> 4;            // 0..15
  const int brow1 = brow0 + 16;          // 16..31
  const int bcol4 = (tid & 15) * 4;      // 0..60

  const float* gA  = x  + (size_t)gra * IN_DIM + acol4;            // += KC per chunk
  const float* gB0 = Wm + (size_t)brow0 * OUTF + n0 + bcol4;       // += KC rows per chunk
  const float* gB1 = Wm + (size_t)brow1 * OUTF + n0 + bcol4;

  // LDS byte addresses (generic LDS pointer: low 32 bits = LDS offset)
  unsigned ldsA[2], ldsB0[2], ldsB1[2];
  #pragma unroll
  for (int b = 0; b < 2; ++b) {
    ldsA[b]  = (unsigned)(unsigned long long)&As[b][arow * AS_STRIDE + acol4];
    ldsB0[b] = (unsigned)(unsigned long long)&Bs[b][brow0 * BS_STRIDE + bcol4];
    ldsB1[b] = (unsigned)(unsigned long long)&Bs[b][brow1 * BS_STRIDE + bcol4];
  }

  v8f c = {};

  // prologue: stream chunk 0 into buffer 0
  ASYNC_B128(ldsA[0],  gA);
  ASYNC_B128(ldsB0[0], gB0);
  ASYNC_B128(ldsB1[0], gB1);

  for (int i = 0; i < NCHUNK; ++i) {
    const int cb = i & 1;
    if (i + 1 < NCHUNK) {
      const int nb = (i + 1) & 1;
      const float* a  = gA  + (size_t)(i + 1) * KC;
      const float* b0 = gB0 + (size_t)(i + 1) * KC * OUTF;
      const float* b1 = gB1 + (size_t)(i + 1) * KC * OUTF;
      ASYNC_B128(ldsA[nb],  a);
      ASYNC_B128(ldsB0[nb], b0);
      ASYNC_B128(ldsB1[nb], b1);
      // 3 newer outstanding allowed -> chunk i's 3 transfers are complete
      asm volatile("s_wait_asynccnt 0x3" ::: "memory");
    } else {
      asm volatile("s_wait_asynccnt 0x0" ::: "memory");
    }
    __syncthreads();

    #pragma unroll
    for (int kk = 0; kk < KC; kk += 4) {
      const int kb = kk + half * 2;
      v2f a, b;
      // A 16x4: lanes0-15 hold K={0,1}, lanes16-31 K={2,3}, M = lane&15
      a.x = As[cb][(wr * 16 + l16) * AS_STRIDE + kb];
      a.y = As[cb][(wr * 16 + l16) * AS_STRIDE + kb + 1];
      // B 4x16: same K split, N = lane&15
      b.x = Bs[cb][kb * BS_STRIDE + wc * 16 + l16];
      b.y = Bs[cb][(kb + 1) * BS_STRIDE + wc * 16 + l16];
      c = __builtin_amdgcn_wmma_f32_16x16x4_f32(
              /*neg_a=*/false, a, /*neg_b=*/false, b,
              /*c_mod=*/(short)0, c, /*reuse_a=*/false, /*reuse_b=*/false);
    }
    __syncthreads();   // all waves done reading buffer cb before it is refilled
  }

  // C/D layout: VGPR r -> M = r + 8*half ; N = lane&15
  #pragma unroll
  for (int r = 0; r < 8; ++r) {
    int row = m0 + wr * 16 + half * 8 + r;
    int col = n0 + wc * 16 + l16;
    if (row < M) outp[(size_t)row * OUTF + col] = c[r];
  }
}

// ---- per-node attention logits: a_src[n,h] = <h[n,h,:], att_src[h,:]> ------
__global__ __launch_bounds__(256) void gat_att_scores(
    const float* __restrict__ hbuf, const float* __restrict__ att_src,
    const float* __restrict__ att_dst, float* __restrict__ a_src,
    float* __restrict__ a_dst, int N)
{
  int wave = threadIdx.x >> 5;
  int lane = threadIdx.x & 31;
  int n = blockIdx.x * 8 + wave;
  if (n >= N) return;
  int head = lane >> 3;            // 8 lanes per head
  int sub  = lane & 7;
  int colb = head * 64 + sub * 8;  // 8 contiguous cols per lane
  const float* hp  = hbuf + (size_t)n * OUTF + colb;
  const float* asv = att_src + colb;
  const float* adv = att_dst + colb;
  float s = 0.f, d = 0.f;
  #pragma unroll
  for (int j = 0; j < 8; ++j) {
    float hv = hp[j];
    s += hv * asv[j];
    d += hv * adv[j];
  }
  #pragma unroll
  for (int m = 1; m < 8; m <<= 1) {
    s += __shfl_xor(s, m, 32);
    d += __shfl_xor(d, m, 32);
  }
  if (sub == 0) {
    a_src[n * HEADS + head] = s;
    a_dst[n * HEADS + head] = d;
  }
}

__device__ __forceinline__ void atomicMaxF(float* addr, float val) {
  // sign-aware int/uint monotone mapping of IEEE754
  if (val >= 0.0f) atomicMax((int*)addr, __float_as_int(val));
  else             atomicMin((unsigned int*)addr, __float_as_uint(val));
}

// ---- self-loop logit seeds running max ------------------------------------
__global__ void gat_self_init(const float* __restrict__ a_src,
                              const float* __restrict__ a_dst,
                              float* __restrict__ selfe,
                              float* __restrict__ emax, int NH)
{
  int i = blockIdx.x * blockDim.x + threadIdx.x;
  if (i >= NH) return;
  float v = a_src[i] + a_dst[i];
  v = v > 0.f ? v : NEG_SLOPE * v;
  selfe[i] = v;
  emax[i]  = v;
}

// ---- edge pass 1: segment max ---------------------------------------------
__global__ void gat_edge_max(const int* __restrict__ ei,
                             const float* __restrict__ a_src,
                             const float* __restrict__ a_dst,
                             float* __restrict__ emax, int E)
{
  int e = blockIdx.x * blockDim.x + threadIdx.x;
  if (e >= E) return;
  int s = ei[e], d = ei[E + e];
  #pragma unroll
  for (int h = 0; h < HEADS; ++h) {
    float v = a_src[s * HEADS + h] + a_dst[d * HEADS + h];
    v = v > 0.f ? v : NEG_SLOPE * v;
    atomicMaxF(&emax[d * HEADS + h], v);
  }
}

// ---- seed agg/denom with the self-loop contribution -----------------------
__global__ void gat_seed(const float* __restrict__ hbuf,
                         const float* __restrict__ selfe,
                         const float* __restrict__ emax,
                         float* __restrict__ agg,
                         float* __restrict__ denom, int N)
{
  size_t i = (size_t)blockIdx.x * blockDim.x + threadIdx.x;
  if (i >= (size_t)N * OUTF) return;
  int n = (int)(i >> 8), col = (int)(i & 255), head = col >> 6;
  float w = __expf(selfe[n * HEADS + head] - emax[n * HEADS + head]);
  agg[i] = w * hbuf[i];
  if ((col & 63) == 0) denom[n * HEADS + head] = w;
}

// ---- edge pass 2: weighted scatter-add (L2-resident atomics) --------------
__global__ __launch_bounds__(256) void gat_edge_agg(
    const int* __restrict__ ei, const float* __restrict__ hbuf,
    const float* __restrict__ a_src, const float* __restrict__ a_dst,
    const float* __restrict__ emax, float* __restrict__ agg,
    float* __restrict__ denom, int E)
{
  int e   = blockIdx.x;
  int col = threadIdx.x;
  int s = ei[e], d = ei[E + e];
  int head = col >> 6;
  float v = a_src[s * HEADS + head] + a_dst[d * HEADS + head];
  v = v > 0.f ? v : NEG_SLOPE * v;
  float w = __expf(v - emax[d * HEADS + head]);
  atomicAdd(&agg[(size_t)d * OUTF + col], w * hbuf[(size_t)s * OUTF + col]);
  if ((col & 63) == 0) atomicAdd(&denom[d * HEADS + head], w);
}

// ---- normalize + bias + skip + ELU (in place in d_out) --------------------
__global__ void gat_final(const float* __restrict__ skip,
                          const float* __restrict__ denom,
                          const float* __restrict__ bias,
                          float* __restrict__ io, int N)
{
  size_t i = (size_t)blockIdx.x * blockDim.x + threadIdx.x;
  if (i >= (size_t)N * OUTF) return;
  int n = (int)(i >> 8), col = (int)(i & 255), head = col >> 6;
  float v = io[i] / (denom[n * HEADS + head] + EPS_) + bias[col] + skip[i];
  io[i] = v > 0.f ? v : expm1f(v);
}

// ---------------------------------------------------------------------------
extern "C" void kernel_launch(void* const* d_in, const int* in_sizes, int n_in,
                              void* d_out, int out_size, void* d_ws, size_t ws_size,
                              hipStream_t stream)
{
  const float* x       = (const float*)d_in[0];
  const int*   ei      = (const int*)d_in[1];     // edge_index [2,E] int32
  const float* W       = (const float*)d_in[2];
  const float* att_src = (const float*)d_in[3];
  const float* att_dst = (const float*)d_in[4];
  const float* bias    = (const float*)d_in[5];
  const float* skip_W  = (const float*)d_in[6];

  const int N = in_sizes[0] / IN_DIM;
  const int E = in_sizes[1] / 2;

  float* ws    = (float*)d_ws;
  float* hbuf  = ws;                                  // [N,256]
  float* skip  = hbuf  + (size_t)N * OUTF;            // [N,256]
  float* a_src = skip  + (size_t)N * OUTF;            // [N,4]
  float* a_dst = a_src + (size_t)N * HEADS;
  float* selfe = a_dst + (size_t)N * HEADS;
  float* emax  = selfe + (size_t)N * HEADS;
  float* denom = emax  + (size_t)N * HEADS;
  float* agg   = (float*)d_out;                       // [N,256] accumulates in d_out

  dim3 gGemm((N + BM - 1) / BM, OUTF / BN, 2);
  gat_gemm_wmma<<<gGemm, 256, 0, stream>>>(x, W, skip_W, hbuf, skip, N);

  gat_att_scores<<<(N + 7) / 8, 256, 0, stream>>>(hbuf, att_src, att_dst,
                                                  a_src, a_dst, N);

  gat_self_init<<<(N * HEADS + 255) / 256, 256, 0, stream>>>(a_src, a_dst,
                                                             selfe, emax, N * HEADS);

  gat_edge_max<<<(E + 255) / 256, 256, 0, stream>>>(ei, a_src, a_dst, emax, E);

  unsigned nelem_blocks = (unsigned)(((size_t)N * OUTF + 255) / 256);
  gat_seed<<<nelem_blocks, 256, 0, stream>>>(hbuf, selfe, emax, agg, denom, N);

  gat_edge_agg<<<(unsigned)E, 256, 0, stream>>>(ei, hbuf, a_src, a_dst,
                                                emax, agg, denom, E);

  gat_final<<<nelem_blocks, 256, 0, stream>>>(skip, denom, bias, agg, N);
}